// slater_det_69939247448350
// MI455X (gfx1250) — compile-verified
//
#include <hip/hip_runtime.h>
#include <hip/hip_bf16.h>
#include <math.h>

// MI455X (gfx1250) implementation of the BdG log-amplitude:
//   logdet(U[occ,:Ne]) + coherent terms + Jastrow quadratic forms
// Dominant cost: 2048x2048 fp32 eigensolve -> blocked two-sided Jacobi with
// V_WMMA_F32_16X16X4_F32 rotation application; rotation tiles staged into LDS
// by the Tensor Data Mover (tensor_load_to_lds + s_wait_tensorcnt);
// 1024x1024 LU with WMMA trailing updates; circulant quadratic forms
// regenerated from 289-entry tables.

#define L1 1024
#define N2 2048
#define NBLK 64          // 64 block-columns of width 32 over N2
#define PW 64            // pair width (2 blocks of 32)
#define NPAIR 32         // pairs per tournament round
#define NROUND 63        // rounds per sweep
#define NSWEEP 4         // global Jacobi sweeps
#define INSWEEP 8        // inner scalar-Jacobi sweeps per 64x64 pair solve

typedef __attribute__((ext_vector_type(2))) float v2f;
typedef __attribute__((ext_vector_type(8))) float v8f;
typedef __attribute__((ext_vector_type(4))) unsigned int u32x4;
typedef __attribute__((ext_vector_type(8))) int i32x8;
typedef __attribute__((ext_vector_type(4))) int i32x4;

// ------------------- Tensor Data Mover descriptor helpers ------------------
// D# per CDNA5 ISA ch.8: group0 = {count/flags, lds_addr, global_addr, type},
// group1 = {mask/data_size/pad, dims/tiles/strides}. This toolchain exposes
// the 6-arg builtin: (g0 u32x4, g1 i32x8, g2 i32x4, g3 i32x4, extra i32x8, cpol).
__device__ __forceinline__ void tdm_load_2d(const void* gptr, void* lptr,
                                            unsigned w_el, unsigned h,
                                            unsigned row_stride_el, int pad1) {
    unsigned long long ga = (unsigned long long)gptr;
    unsigned lds = (unsigned)(unsigned long long)lptr;   // low 32b of flat = LDS offset
    u32x4 g0 = {};
    i32x8 g1 = {};
    i32x4 g2 = {}, g3 = {};
    i32x8 g4 = {};
    g0[0] = 1u;                                          // count=1, user mode
    g0[1] = lds;                                         // lds_addr (bytes)
    g0[2] = (unsigned)(ga & 0xFFFFFFFFu);                // global_addr[31:0]
    g0[3] = (unsigned)((ga >> 32) & 0x01FFFFFFu) | (2u << 30);  // addr[56:32] | type=2
    unsigned dw0 = (2u << 16);                           // data_size = 4 bytes
    if (pad1) dw0 |= (1u << 20) | (5u << 22);            // pad_enable, interval=64 DW, amount=1 DW
    g1[0] = (int)dw0;
    g1[1] = (int)((row_stride_el & 0xFFFFu) << 16);      // tensor_dim0[15:0] @ bit48
    g1[2] = (int)(((row_stride_el >> 16) & 0xFFFFu)      // tensor_dim0[31:16]
                  | ((h & 0xFFFFu) << 16));              // tensor_dim1[15:0] @ bit80
    g1[3] = (int)(((h >> 16) & 0xFFFFu)                  // tensor_dim1[31:16]
                  | ((w_el & 0xFFFFu) << 16));           // tile_dim0 @ bit112
    g1[4] = (int)(h > 1 ? (h & 0xFFFFu) : 0u);           // tile_dim1 (0 = unused)
    g1[5] = (int)row_stride_el;                          // tensor_dim0_stride[31:0]
    g1[6] = 0;                                           // stride high bits / dim1_stride
    g1[7] = 0;
    __builtin_amdgcn_tensor_load_to_lds(g0, g1, g2, g3, g4, 0);
}

__device__ __forceinline__ void tdm_load_1d(const void* gptr, void* lptr, unsigned n_el) {
    tdm_load_2d(gptr, lptr, n_el, 1, n_el, 0);
}

// ---------------- tournament schedule on 64 block indices -----------------
__device__ __forceinline__ void pair_pq(int r, int t, int& P, int& Q) {
    if (t == 0) { P = NBLK - 1; Q = r % (NBLK - 1); }
    else        { P = (r + t) % (NBLK - 1); Q = (r + NBLK - 1 - t) % (NBLK - 1); }
    if (P > Q) { int tmp = P; P = Q; Q = tmp; }
}

// circulant lookup: V(dx,dy) from a (1+Lx/2)*(1+Ly/2)=289 table (flat 288/287)
__device__ __forceinline__ float circv(const float* flat, int pad_front, int dx, int dy) {
    int mx = (dx <= 16) ? dx : 32 - dx;
    int my = (dy <= 16) ? dy : 32 - dy;
    int k = mx * 17 + my;
    if (pad_front) return (k >= 1 && k <= 287) ? flat[k - 1] : 0.0f;
    return (k < 288) ? flat[k] : 0.0f;
}

// ---------------------------- setup kernels -------------------------------
__global__ void zero_acc_k(float* acc) { if (threadIdx.x < 8) acc[threadIdx.x] = 0.0f; }

__global__ void build_h0_k(const float* __restrict__ hop, const float* __restrict__ sw,
                           const float* __restrict__ dw, const float* __restrict__ hv,
                           const float* __restrict__ sv, const float* __restrict__ dv,
                           float* __restrict__ A) {
    long long idx = (long long)blockIdx.x * 256 + threadIdx.x;
    const long long NS = (long long)N2 * N2;
    if (idx >= NS) return;
    float v = -hop[NS + idx] + hv[0] * hop[idx] + hv[1] * hop[2 * NS + idx]
              + sv[0] * sw[idx] + dv[0] * dw[idx];
    A[idx] = v;
}

__global__ void phonon_scatter_k(float* __restrict__ A, const int* __restrict__ ivic,
                                 const float* __restrict__ X, const float* __restrict__ Y,
                                 const float* __restrict__ sign_b,
                                 const float* g, const float* fS, const float* fD) {
    int t = blockIdx.x * blockDim.x + threadIdx.x;
    if (t >= L1 * 4) return;
    int i = t >> 2, d = t & 3;
    int j = ivic[i * 4 + d];
    bool isx = (d == 1) || (d == 3);            // x-neighbors feed phonX
    float s = sign_b[(long long)i * L1 + j];
    float val = isx ? (X[j] - X[i]) * s : (Y[j] - Y[i]) * s;
    float gP  = g[0] * val;
    float fsP = fS[0] * val;
    float fdM = fD[0] * (isx ? val : -val);     // M = phonX - phonY
    long long ld = N2;
    A[(long long)i * ld + j]                += gP;
    A[(long long)(i + L1) * ld + (j + L1)]  -= gP;
    A[(long long)i * ld + (j + L1)]         += fsP + fdM;
    A[(long long)(i + L1) * ld + j]         += fsP + fdM;
}

__global__ void init_identity_k(float* __restrict__ U) {
    long long idx = (long long)blockIdx.x * 256 + threadIdx.x;
    if (idx >= (long long)N2 * N2) return;
    int i = (int)(idx / N2), j = (int)(idx % N2);
    U[idx] = (i == j) ? 1.0f : 0.0f;
}

// ---------------- 64x64 pair eigensolve (scalar Jacobi in LDS) ------------
__global__ void pair_solve_k(const float* __restrict__ A, float* __restrict__ Vb, int r) {
    __shared__ float S[PW][PW + 1];
    __shared__ float V[PW][PW + 1];
    __shared__ float cs0[NPAIR];
    __shared__ float cs1[NPAIR];
    int t = blockIdx.x;          // pair id
    int tid = threadIdx.x;       // 64 threads
    int P, Q; pair_pq(r, t, P, Q);
    for (int e = tid; e < PW * PW; e += 64) {
        int li = e >> 6, lj = e & 63;
        int gi = (li < 32) ? (P * 32 + li) : (Q * 32 + li - 32);
        int gj = (lj < 32) ? (P * 32 + lj) : (Q * 32 + lj - 32);
        S[li][lj] = A[(long long)gi * N2 + gj];
        V[li][lj] = (li == lj) ? 1.0f : 0.0f;
    }
    __syncthreads();
    for (int sweep = 0; sweep < INSWEEP; ++sweep) {
        for (int rr = 0; rr < PW - 1; ++rr) {
            if (tid < NPAIR) {
                int p, q;
                if (tid == 0) { p = PW - 1; q = rr % (PW - 1); }
                else          { p = (rr + tid) % (PW - 1); q = (rr + PW - 1 - tid) % (PW - 1); }
                if (p > q) { int tmp = p; p = q; q = tmp; }
                float app = S[p][p], aqq = S[q][q], apq = S[p][q];
                float c = 1.0f, s = 0.0f;
                if (fabsf(apq) > 1e-12f) {
                    float tau = (aqq - app) / (2.0f * apq);
                    float tt = (tau >= 0.0f ? 1.0f : -1.0f) / (fabsf(tau) + sqrtf(1.0f + tau * tau));
                    c = rsqrtf(1.0f + tt * tt);
                    s = tt * c;
                }
                cs0[tid] = c; cs1[tid] = s;
            }
            __syncthreads();
            // column pass on S and V (pairs are disjoint)
            for (int w = tid; w < NPAIR * PW; w += 64) {
                int pr = w >> 6, i = w & 63;
                int p, q;
                if (pr == 0) { p = PW - 1; q = rr % (PW - 1); }
                else         { p = (rr + pr) % (PW - 1); q = (rr + PW - 1 - pr) % (PW - 1); }
                if (p > q) { int tmp = p; p = q; q = tmp; }
                float c = cs0[pr], s = cs1[pr];
                float sp = S[i][p], sq = S[i][q];
                S[i][p] = c * sp - s * sq; S[i][q] = s * sp + c * sq;
                float vp = V[i][p], vq = V[i][q];
                V[i][p] = c * vp - s * vq; V[i][q] = s * vp + c * vq;
            }
            __syncthreads();
            // row pass on S
            for (int w = tid; w < NPAIR * PW; w += 64) {
                int pr = w >> 6, i = w & 63;
                int p, q;
                if (pr == 0) { p = PW - 1; q = rr % (PW - 1); }
                else         { p = (rr + pr) % (PW - 1); q = (rr + PW - 1 - pr) % (PW - 1); }
                if (p > q) { int tmp = p; p = q; q = tmp; }
                float c = cs0[pr], s = cs1[pr];
                float sp = S[p][i], sq = S[q][i];
                S[p][i] = c * sp - s * sq; S[q][i] = s * sp + c * sq;
            }
            __syncthreads();
        }
    }
    float* Vo = Vb + (long long)t * PW * PW;
    for (int e = tid; e < PW * PW; e += 64) Vo[e] = V[e >> 6][e & 63];
}

// ---- in-place column rotation: M[:, pair cols] <- M[:, pair cols] * V_t ----
// grid (128 row-strips, 32 pairs), 128 threads = 4 waves, one 16x16 tile/wave.
// V_t (16KB) is staged LDS-side by the Tensor Data Mover and shared by all 4
// waves; the WMMA inner loop then reads only LDS.
__global__ void colrot_k(float* __restrict__ M, const float* __restrict__ Vb, int r) {
    __shared__ float s[16][PW + 4];
    __shared__ float sV[PW][PW];
    int strip = blockIdx.x, t = blockIdx.y;
    int P, Q; pair_pq(r, t, P, Q);
    int tid = threadIdx.x;
    const float* V = Vb + (long long)t * PW * PW;
    if (tid < 32) tdm_load_1d(V, &sV[0][0], PW * PW);    // wave0 kicks the DMA
    int rowbase = strip * 16;
    for (int e = tid; e < 16 * PW; e += 128) {
        int rr = e >> 6, l = e & 63;
        int gj = (l < 32) ? (P * 32 + l) : (Q * 32 + l - 32);
        s[rr][l] = M[(long long)(rowbase + rr) * N2 + gj];
    }
    if (tid < 32) __builtin_amdgcn_s_wait_tensorcnt(0);
    __syncthreads();
    int w = tid >> 5, lane = tid & 31, lo = lane & 15, hi = lane >> 4;
    v8f d = {};
    for (int k0 = 0; k0 < PW; k0 += 4) {
        v2f a, b;
        a.x = s[lo][k0 + 2 * hi];
        a.y = s[lo][k0 + 2 * hi + 1];
        b.x = sV[k0 + 2 * hi][16 * w + lo];
        b.y = sV[k0 + 2 * hi + 1][16 * w + lo];
        d = __builtin_amdgcn_wmma_f32_16x16x4_f32(false, a, false, b, (short)0, d, false, false);
    }
    int lcol = 16 * w + lo;
    int gj = (lcol < 32) ? (P * 32 + lcol) : (Q * 32 + lcol - 32);
#pragma unroll
    for (int v = 0; v < 8; ++v)
        M[(long long)(rowbase + v + 8 * hi) * N2 + gj] = d[v];
}

// ---- in-place row rotation: M[pair rows, :] <- V_t^T * M[pair rows, :] ----
__global__ void rowrot_k(float* __restrict__ M, const float* __restrict__ Vb, int r) {
    __shared__ float s[PW][16 + 1];
    __shared__ float sV[PW][PW];
    int strip = blockIdx.x, t = blockIdx.y;
    int P, Q; pair_pq(r, t, P, Q);
    int tid = threadIdx.x;
    const float* V = Vb + (long long)t * PW * PW;
    if (tid < 32) tdm_load_1d(V, &sV[0][0], PW * PW);
    int colbase = strip * 16;
    for (int e = tid; e < PW * 16; e += 128) {
        int l = e >> 4, cc = e & 15;
        int gi = (l < 32) ? (P * 32 + l) : (Q * 32 + l - 32);
        s[l][cc] = M[(long long)gi * N2 + colbase + cc];
    }
    if (tid < 32) __builtin_amdgcn_s_wait_tensorcnt(0);
    __syncthreads();
    int w = tid >> 5, lane = tid & 31, lo = lane & 15, hi = lane >> 4;
    v8f d = {};
    for (int k0 = 0; k0 < PW; k0 += 4) {
        v2f a, b;
        a.x = sV[k0 + 2 * hi][16 * w + lo];              // A[m][k] = V[k][16w+m]
        a.y = sV[k0 + 2 * hi + 1][16 * w + lo];
        b.x = s[k0 + 2 * hi][lo];
        b.y = s[k0 + 2 * hi + 1][lo];
        d = __builtin_amdgcn_wmma_f32_16x16x4_f32(false, a, false, b, (short)0, d, false, false);
    }
#pragma unroll
    for (int v = 0; v < 8; ++v) {
        int m = 16 * w + v + 8 * hi;
        int gi = (m < 32) ? (P * 32 + m) : (Q * 32 + m - 32);
        M[(long long)gi * N2 + colbase + lo] = d[v];
    }
}

// ---------------- eigen-column selection (lowest 1024) --------------------
__global__ void extract_eig_k(const float* __restrict__ A, float* __restrict__ eig) {
    int j = blockIdx.x * 256 + threadIdx.x;
    if (j < N2) eig[j] = A[(long long)j * N2 + j];
}

__global__ void rank_k(const float* __restrict__ eig, int* __restrict__ rank) {
    __shared__ int red[256];
    int j = blockIdx.x;
    float e = eig[j];
    int cnt = 0;
    for (int i = threadIdx.x; i < N2; i += 256) {
        float ei = eig[i];
        cnt += (ei < e) || (ei == e && i < j);
    }
    red[threadIdx.x] = cnt;
    __syncthreads();
    for (int s2 = 128; s2 > 0; s2 >>= 1) {
        if (threadIdx.x < s2) red[threadIdx.x] += red[threadIdx.x + s2];
        __syncthreads();
    }
    if (threadIdx.x == 0) rank[j] = red[0];
}

__global__ void gather_W_k(const float* __restrict__ U, const int* __restrict__ rank,
                           const int* __restrict__ occ, float* __restrict__ Wm) {
    int j = blockIdx.x;
    int rk = rank[j];
    if (rk >= L1) return;
    for (int i = threadIdx.x; i < L1; i += 256)
        Wm[(long long)i * L1 + rk] = U[(long long)occ[i] * N2 + j];
}

// ----------------------- blocked LU for slogdet ---------------------------
__global__ void lu_panel_k(float* __restrict__ Wm, int kb, float* __restrict__ acc) {
    __shared__ float prow[64];
    __shared__ float pinv_s;
    __shared__ float logsum;
    int tid = threadIdx.x;
    if (tid == 0) logsum = 0.0f;
    __syncthreads();
    for (int kk = 0; kk < 64; ++kk) {
        int pr = kb + kk;
        if (tid == 0) {
            float piv = Wm[(long long)pr * L1 + pr];
            pinv_s = 1.0f / piv;
            logsum += logf(fabsf(piv) + 1e-30f);
        }
        __syncthreads();
        for (int c = tid; c < 64; c += 256) prow[c] = Wm[(long long)pr * L1 + kb + c];
        __syncthreads();
        float pinv = pinv_s;
        for (int i = pr + 1 + tid; i < L1; i += 256) {
            long long ro = (long long)i * L1 + kb;
            float lik = Wm[ro + kk] * pinv;
            Wm[ro + kk] = lik;
            for (int c = kk + 1; c < 64; ++c) Wm[ro + c] -= lik * prow[c];
        }
        __syncthreads();
    }
    if (tid == 0) atomicAdd(acc, logsum);
}

// L11 (64x64, row stride 1024) pulled by the TDM as a 2D tile directly into
// the padded 64x65 LDS layout (pad_interval=64 DW, pad_amount=1 DW).
__global__ void lu_trsm_k(float* __restrict__ Wm, int kb) {
    __shared__ float Lm[64][65];
    int tid = threadIdx.x;   // 64
    int j = kb + 64 + blockIdx.x * 64 + tid;
    if (tid < 32) {
        tdm_load_2d(&Wm[(long long)kb * L1 + kb], &Lm[0][0], 64, 64, L1, 1);
        __builtin_amdgcn_s_wait_tensorcnt(0);
    }
    __syncthreads();
    float x[64];
    for (int i = 0; i < 64; ++i) {
        float a = Wm[(long long)(kb + i) * L1 + j];
        for (int t2 = 0; t2 < i; ++t2) a -= Lm[i][t2] * x[t2];
        x[i] = a;
        Wm[(long long)(kb + i) * L1 + j] = a;
    }
}

// trailing update C -= L21 * U12 via fp32 WMMA (one 16x16 tile per wave)
__global__ void lu_gemm_k(float* __restrict__ Wm, int kb) {
    int lane = threadIdx.x, lo = lane & 15, hi = lane >> 4;
    int rb = kb + 64 + blockIdx.x * 16;
    int cb = kb + 64 + blockIdx.y * 16;
    v8f d = {};
    for (int k0 = 0; k0 < 64; k0 += 4) {
        v2f a, b;
        a.x = Wm[(long long)(rb + lo) * L1 + kb + k0 + 2 * hi];
        a.y = Wm[(long long)(rb + lo) * L1 + kb + k0 + 2 * hi + 1];
        b.x = Wm[(long long)(kb + k0 + 2 * hi) * L1 + cb + lo];
        b.y = Wm[(long long)(kb + k0 + 2 * hi + 1) * L1 + cb + lo];
        d = __builtin_amdgcn_wmma_f32_16x16x4_f32(false, a, false, b, (short)0, d, false, false);
    }
#pragma unroll
    for (int v = 0; v < 8; ++v) {
        long long off = (long long)(rb + v + 8 * hi) * L1 + cb + lo;
        Wm[off] -= d[v];
    }
}

// -------- fused Jastrow quadratic forms (circulants from tables) ----------
__global__ void jastrow_all_k(const float* __restrict__ Sz, const float* __restrict__ X,
                              const float* __restrict__ Y, const float* __restrict__ Jspin,
                              const float* __restrict__ JXel, const float* __restrict__ JYel,
                              const float* __restrict__ JXX, const float* __restrict__ JXY,
                              const float* __restrict__ JYY, const float* __restrict__ sign_b,
                              float* __restrict__ acc) {
    __shared__ float red[256];
    int i = blockIdx.x;
    int xi = i >> 5, yi = i & 31;
    float Si = Sz[i], Xi = X[i], Yi = Y[i];
    float sum = 0.0f;
    for (int j = threadIdx.x; j < L1; j += 256) {
        int xj = j >> 5, yj = j & 31;
        int dx = (xj - xi) & 31, dy = (yj - yi) & 31;
        float Sj = Sz[j], Xj = X[j], Yj = Y[j];
        sum += Si * Sj * circv(Jspin, 0, dx, dy);
        sum += Xi * Xj * circv(JXX, 0, dx, dy);
        sum += circv(JXY, 0, dx, dy) * (Yi * Xj + Xi * Yj);
        sum += Yi * Yj * circv(JYY, 0, dx, dy);
        float sb = sign_b[(long long)i * L1 + j];
        float ss = Si * Sj * sb;
        sum += ss * circv(JXel, 1, dx, dy) * (Xi - Xj);
        sum += ss * circv(JYel, 1, dx, dy) * (Yi - Yj);
    }
    red[threadIdx.x] = sum;
    __syncthreads();
    for (int s2 = 128; s2 > 0; s2 >>= 1) {
        if (threadIdx.x < s2) red[threadIdx.x] += red[threadIdx.x + s2];
        __syncthreads();
    }
    if (threadIdx.x == 0) atomicAdd(acc + 1, red[0]);
}

__global__ void coh_base_k(const float* __restrict__ X, const float* __restrict__ Y,
                           const float* __restrict__ sx, const float* __restrict__ sy,
                           const float* zx, const float* zy, const float* xr,
                           const float* yr, float* __restrict__ acc) {
    __shared__ float red[256];
    float sum = 0.0f;
    for (int i = threadIdx.x; i < L1; i += 256) {
        float dxp = X[i] - zx[0] * sx[i];
        float dyp = Y[i] - zy[0] * sy[i];
        sum += -0.5f * xr[0] * dxp * dxp - 0.5f * yr[0] * dyp * dyp;
    }
    red[threadIdx.x] = sum;
    __syncthreads();
    for (int s2 = 128; s2 > 0; s2 >>= 1) {
        if (threadIdx.x < s2) red[threadIdx.x] += red[threadIdx.x + s2];
        __syncthreads();
    }
    if (threadIdx.x == 0) atomicAdd(acc + 1, red[0]);
}

__global__ void finalize_k(const float* __restrict__ acc, float* __restrict__ out) {
    if (threadIdx.x == 0) out[0] = acc[0] + acc[1];
}

// ------------------------------- host side --------------------------------
extern "C" void kernel_launch(void* const* d_in, const int* in_sizes, int n_in,
                              void* d_out, int out_size, void* d_ws, size_t ws_size,
                              hipStream_t stream) {
    (void)in_sizes; (void)n_in; (void)out_size; (void)ws_size;
    const long long NS = (long long)N2 * N2;

    float* ws   = (float*)d_ws;
    float* A    = ws;                                  // 2048^2
    float* U    = ws + NS;                             // 2048^2
    float* Vb   = U + NS;                              // 32 * 64 * 64
    float* Wm   = Vb + (long long)NPAIR * PW * PW;     // 1024^2
    float* eig  = Wm + (long long)L1 * L1;             // 2048
    int*   rank = (int*)(eig + N2);                    // 2048
    float* acc  = (float*)(rank + N2);                 // 8

    const int*   occ   = (const int*)d_in[0];
    const float* Sz    = (const float*)d_in[2];
    const float* X     = (const float*)d_in[3];
    const float* Y     = (const float*)d_in[4];
    const float* hv    = (const float*)d_in[5];
    const float* sv    = (const float*)d_in[6];
    const float* dv    = (const float*)d_in[7];
    const float* Jspin = (const float*)d_in[8];
    const float* JXel  = (const float*)d_in[9];
    const float* JYel  = (const float*)d_in[10];
    const float* JXX   = (const float*)d_in[11];
    const float* JXY   = (const float*)d_in[12];
    const float* JYY   = (const float*)d_in[13];
    const float* g     = (const float*)d_in[14];
    const float* fS    = (const float*)d_in[15];
    const float* fD    = (const float*)d_in[16];
    const float* zx    = (const float*)d_in[17];
    const float* zy    = (const float*)d_in[18];
    const float* xr    = (const float*)d_in[19];
    const float* yr    = (const float*)d_in[20];
    const float* hopm  = (const float*)d_in[21];
    const float* swm   = (const float*)d_in[22];
    const float* dwm   = (const float*)d_in[23];
    const int*   ivic  = (const int*)d_in[24];
    const float* signb = (const float*)d_in[25];
    const float* sx    = (const float*)d_in[26];
    const float* sy    = (const float*)d_in[27];

    zero_acc_k<<<1, 32, 0, stream>>>(acc);
    build_h0_k<<<(int)((NS + 255) / 256), 256, 0, stream>>>(hopm, swm, dwm, hv, sv, dv, A);
    phonon_scatter_k<<<(L1 * 4 + 255) / 256, 256, 0, stream>>>(A, ivic, X, Y, signb, g, fS, fD);
    init_identity_k<<<(int)((NS + 255) / 256), 256, 0, stream>>>(U);

    for (int sweep = 0; sweep < NSWEEP; ++sweep) {
        for (int r = 0; r < NROUND; ++r) {
            pair_solve_k<<<NPAIR, 64, 0, stream>>>(A, Vb, r);
            colrot_k<<<dim3(N2 / 16, NPAIR), 128, 0, stream>>>(A, Vb, r);
            rowrot_k<<<dim3(N2 / 16, NPAIR), 128, 0, stream>>>(A, Vb, r);
            colrot_k<<<dim3(N2 / 16, NPAIR), 128, 0, stream>>>(U, Vb, r);
        }
    }

    extract_eig_k<<<N2 / 256, 256, 0, stream>>>(A, eig);
    rank_k<<<N2, 256, 0, stream>>>(eig, rank);
    gather_W_k<<<N2, 256, 0, stream>>>(U, rank, occ, Wm);

    for (int kb = 0; kb < L1; kb += 64) {
        lu_panel_k<<<1, 256, 0, stream>>>(Wm, kb, acc);
        int rest = L1 - kb - 64;
        if (rest > 0) {
            lu_trsm_k<<<rest / 64, 64, 0, stream>>>(Wm, kb);
            lu_gemm_k<<<dim3(rest / 16, rest / 16), 32, 0, stream>>>(Wm, kb);
        }
    }

    jastrow_all_k<<<L1, 256, 0, stream>>>(Sz, X, Y, Jspin, JXel, JYel, JXX, JXY, JYY, signb, acc);
    coh_base_k<<<1, 256, 0, stream>>>(X, Y, sx, sy, zx, zy, xr, yr, acc);
    finalize_k<<<1, 32, 0, stream>>>(acc, (float*)d_out);
}